// Net_2078764171470
// MI455X (gfx1250) — compile-verified
//
#include <hip/hip_runtime.h>

typedef __attribute__((ext_vector_type(2))) float v2f;
typedef __attribute__((ext_vector_type(8))) float v8f;

__device__ __forceinline__ float bperm_lane(int srclane, float v) {
  return __int_as_float(__builtin_amdgcn_ds_bpermute(srclane << 2, __float_as_int(v)));
}

__device__ __forceinline__ float relu(float v) { return fmaxf(v, 0.f); }

__device__ __forceinline__ v8f wmma_f32(v2f a, v2f b, v8f c) {
  // 8 args: (neg_a, A, neg_b, B, c_mod, C, reuse_a, reuse_b)
  return __builtin_amdgcn_wmma_f32_16x16x4_f32(false, a, false, b, (short)0, c,
                                               false, false);
}

// Transposed MLP per 16-batch tile: Y[16x16] = A(W pad 16x12) x B(X pad 12x16) + C(bias)
// A layout (f32 16x4): lane L, vgpr v of chunk i -> W[m=L&15][k=4i+v+2*(L>>4)]
// B layout (f32 4x16): lane L, vgpr v of chunk i -> X[k=4i+v+2*(L>>4)][n=L&15]
// C/D layout: vgpr j, lane L -> Y[j+8*(L>>4)][L&15]
struct LaneConsts {
  float w1k[6], b1k[6];
  v2f a2[3], a3[3];
  v8f c2, c3;
  float w4p[8];
  float b4s;
  int hi, lo;
};

// One 16-element batch tile through layers 1-4. Returns the final scalar for
// batch index (tile_base + lo); the SWAPX16 fold makes it valid in BOTH lane
// halves. EXEC must be all-ones (no divergence inside).
__device__ __forceinline__ float mlp_tile(float xv, const LaneConsts& K) {
  const int hi = K.hi, lo = K.lo;

  // Layer 1 (outer product) -> B operand chunks for layer 2, in-register.
  // fma results are known-canonical: fmaxf lowers to a single v_max_num.
  v2f B0, B1, B2c;
  B0[0]  = relu(fmaf(K.w1k[0], xv, K.b1k[0]));
  B0[1]  = relu(fmaf(K.w1k[1], xv, K.b1k[1]));
  B1[0]  = relu(fmaf(K.w1k[2], xv, K.b1k[2]));
  B1[1]  = relu(fmaf(K.w1k[3], xv, K.b1k[3]));
  B2c[0] = relu(fmaf(K.w1k[4], xv, K.b1k[4]));
  B2c[1] = relu(fmaf(K.w1k[5], xv, K.b1k[5]));

  // Layer 2: chained K=4 WMMAs, bias as initial C.
  v8f d = wmma_f32(K.a2[0], B0, K.c2);
  d = wmma_f32(K.a2[1], B1, d);
  d = wmma_f32(K.a2[2], B2c, d);

  // D layout -> B layout for layer 3. ReLU commutes with the permute/select,
  // so apply it AFTER conversion: 6 relus instead of 8, and the bpermutes can
  // issue as soon as the WMMA result lands (no relu on the DS critical path).
  // Padded rows 10,11 are zero by construction (A/C padding).
  v2f E0, E1, E2;
  {
    float t0 = bperm_lane(lo, d[2]);       // X[2, n] for upper half
    float t1 = bperm_lane(lo, d[3]);       // X[3, n]
    float t2 = bperm_lane(lo, d[6]);       // X[6, n]
    float t3 = bperm_lane(lo, d[7]);       // X[7, n]
    float u0 = bperm_lane(lo + 16, d[0]);  // X[8, n] for lower half
    float u1 = bperm_lane(lo + 16, d[1]);  // X[9, n]
    E0[0] = relu(hi ? t0 : d[0]);
    E0[1] = relu(hi ? t1 : d[1]);
    E1[0] = relu(hi ? t2 : d[4]);
    E1[1] = relu(hi ? t3 : d[5]);
    E2[0] = relu(hi ? 0.f : u0);           // k=10 row is zero pad
    E2[1] = relu(hi ? 0.f : u1);           // k=11 row is zero pad
  }

  // Layer 3.
  v8f e = wmma_f32(K.a3[0], E0, K.c3);
  e = wmma_f32(K.a3[1], E1, e);
  e = wmma_f32(K.a3[2], E2, e);
#pragma unroll
  for (int j = 0; j < 8; ++j) e[j] = relu(e[j]);

  // Layer 4: two partial chains for ILP, then SWAPX16 fold (total becomes
  // valid in both halves).
  float p0 = 0.f, p1 = 0.f;
#pragma unroll
  for (int j = 0; j < 8; j += 2) {
    p0 = fmaf(K.w4p[j], e[j], p0);
    p1 = fmaf(K.w4p[j + 1], e[j + 1], p1);
  }
  float p = p0 + p1;
  float q = __int_as_float(
      __builtin_amdgcn_ds_swizzle(__float_as_int(p), 0x401f));  // xor 0x10
  return p + q + K.b4s;
}

__global__ __launch_bounds__(256) void mlp4_wmma_kernel(
    const float* __restrict__ x,
    const float* __restrict__ w1, const float* __restrict__ b1,
    const float* __restrict__ w2, const float* __restrict__ b2,
    const float* __restrict__ w3, const float* __restrict__ b3,
    const float* __restrict__ w4, const float* __restrict__ b4,
    float* __restrict__ out, int n)
{
  const int lane = (int)(threadIdx.x & 31u);
  LaneConsts K;
  K.hi = lane >> 4;   // 0: lanes 0-15, 1: lanes 16-31
  K.lo = lane & 15;
  const int hi = K.hi, lo = K.lo;

  // ---------------- per-wave constant setup ----------------
  const int klo_tab[6] = {0, 1, 4, 5, 8, 9};  // k for lanes 0-15; +2 for 16-31
#pragma unroll
  for (int j = 0; j < 6; ++j) {
    int k   = klo_tab[j] + 2 * hi;
    bool ok = (k < 10);
    K.w1k[j] = ok ? w1[k] : 0.f;
    K.b1k[j] = ok ? b1[k] : 0.f;
  }
#pragma unroll
  for (int i = 0; i < 3; ++i) {
#pragma unroll
    for (int v = 0; v < 2; ++v) {
      int m = lo, k = 4 * i + v + 2 * hi;
      bool ok = (m < 10) && (k < 10);
      K.a2[i][v] = ok ? w2[m * 10 + k] : 0.f;
      K.a3[i][v] = ok ? w3[m * 10 + k] : 0.f;
    }
  }
#pragma unroll
  for (int j = 0; j < 8; ++j) {
    int m = j + 8 * hi;
    K.c2[j]  = (m < 10) ? b2[m] : 0.f;
    K.c3[j]  = (m < 10) ? b3[m] : 0.f;
    K.w4p[j] = (m < 10) ? w4[m] : 0.f;
  }
  K.b4s = b4[0];

  // ---------------- grid-stride over PAIRS of 16-element tiles ----------------
  // Each iteration handles 32 consecutive batch elements (one full cacheline):
  // tile A = [base, base+16), tile B = [base+16, base+32).
  const int npairs = (n + 31) >> 5;
  const int wave   = (int)((blockIdx.x * blockDim.x + threadIdx.x) >> 5);
  const int nwaves = (int)((gridDim.x * blockDim.x) >> 5);
  const int nclamp = n - 1;

  for (int pair = wave; pair < npairs; pair += nwaves) {
    const int base = pair << 5;

    // Issue both loads before use; unguarded speculative prefetch of this
    // wave's next line (invalid addresses are silently dropped).
    int ga = base + lo;
    int gb = base + 16 + lo;
    float xa = x[ga < n ? ga : nclamp];
    float xb = x[gb < n ? gb : nclamp];
    __builtin_prefetch(x + base + (nwaves << 5) + lane, 0, 0);

    // Two independent WMMA chains; compiler interleaves them to cover
    // WMMA->WMMA hazards and ds_bpermute latency.
    float totA = mlp_tile(xa, K);
    float totB = mlp_tile(xb, K);

    // lanes 0-15 carry tile A, lanes 16-31 carry tile B -> one 128B store.
    // Output is never re-read: non-temporal store keeps it out of L2.
    float tot = hi ? totB : totA;
    if (base + 32 <= n) {            // wave-uniform fast path: no exec juggling
      __builtin_nontemporal_store(tot, out + base + lane);
    } else {
      int go = base + lane;
      if (go < n) __builtin_nontemporal_store(tot, out + go);
    }
  }
}

extern "C" void kernel_launch(void* const* d_in, const int* in_sizes, int n_in,
                              void* d_out, int out_size, void* d_ws, size_t ws_size,
                              hipStream_t stream) {
  (void)n_in; (void)out_size; (void)d_ws; (void)ws_size;
  const float* x  = (const float*)d_in[0];
  const float* w1 = (const float*)d_in[1];
  const float* b1 = (const float*)d_in[2];
  const float* w2 = (const float*)d_in[3];
  const float* b2 = (const float*)d_in[4];
  const float* w3 = (const float*)d_in[5];
  const float* b3 = (const float*)d_in[6];
  const float* w4 = (const float*)d_in[7];
  const float* b4 = (const float*)d_in[8];
  float* out = (float*)d_out;

  int n      = in_sizes[0];
  int npairs = (n + 31) >> 5;
  long long waves_wanted = npairs;
  int blocks = (int)((waves_wanted * 32 + 255) / 256);
  if (blocks > 1024) blocks = 1024;
  if (blocks < 1) blocks = 1;

  mlp4_wmma_kernel<<<blocks, 256, 0, stream>>>(x, w1, b1, w2, b2, w3, b3, w4, b4,
                                               out, n);
}